// GuidedAttentionLoss_2439541424205
// MI455X (gfx1250) — compile-verified
//
#include <hip/hip_runtime.h>

// GuidedAttentionLoss for MI455X (gfx1250).
// Memory-bound streaming reduction: 256 MiB in, 128 floats out.
// Roofline: 256 MiB / 23.3 TB/s ~= 11.5 us; compute (~67M v_exp_f32) is
// comfortably underneath, so the kernel is organized around HBM streaming:
//  - CDNA5 async global->LDS loads (ASYNCcnt), 4-deep per-wave pipeline
//  - non-temporal (th:TH_LOAD_NT) loads: 256 MiB streamed once > 192 MB L2
//  - wave-private LDS staging => no __syncthreads in the hot loop
//  - deterministic two-pass reduction via d_ws (no float atomics)

namespace {
constexpr int   NB        = 64;     // batches
constexpr int   BLK_PER_B = 32;     // blocks per batch
constexpr int   THREADS   = 256;    // 8 wave32s
constexpr int   WAVES     = THREADS / 32;
constexpr int   TILES     = 32;     // tiles per wave (512 B each)
constexpr int   PIPE      = 4;      // async pipeline depth
constexpr float NEG_INV_2SIG2 = -3.125f;   // -1/(2*0.4^2)
}

__global__ __launch_bounds__(THREADS) void gal_partial_kernel(
    const float* __restrict__ att,
    const int*   __restrict__ ilens,
    const int*   __restrict__ olens,
    float*       __restrict__ partials)
{
  // Wave-private staging: 8 waves * 4 buffers * 32 lanes * 16 B = 16 KiB
  __shared__ float4 smem[WAVES * PIPE * 32];
  __shared__ float  red[2 * WAVES];

  const int blk = blockIdx.x;
  const int b   = blk / BLK_PER_B;
  const int tid = threadIdx.x;
  const int w   = tid >> 5;
  const int l   = tid & 31;

  const int il = ilens[b];
  const int ol = olens[b];
  const float inv_il = 1.0f / (float)il;
  const float inv_ol = 1.0f / (float)ol;

  // Byte offset of this lane's first 16B element within att
  const float*   base  = att;
  const unsigned gOff0 = (unsigned)blk * 131072u          // 128 KiB / block
                       + (unsigned)w   * 16384u           // 16 KiB / wave
                       + (unsigned)l   * 16u;
  // Element index within this batch's 2048x512 matrix
  const unsigned baseElem = (unsigned)(blk % BLK_PER_B) * 32768u
                          + (unsigned)w * 4096u
                          + (unsigned)l * 4u;

  float s1 = 0.0f, s2 = 0.0f;

  auto issue = [&](int t) {
    // LDS byte address of this lane's slot in ring buffer (t mod PIPE).
    // Generic-pointer low 32 bits == workgroup-relative LDS offset.
    unsigned lds = (unsigned)(unsigned long long)
        (&smem[(w * PIPE + (t & (PIPE - 1))) * 32 + l]);
    unsigned go = gOff0 + (unsigned)t * 512u;   // 512 B per wave-tile
    asm volatile("global_load_async_to_lds_b128 %0, %1, %2 th:TH_LOAD_NT"
                 :: "v"(lds), "v"(go), "s"(base)
                 : "memory");
  };

  auto consume = [&](int t) {
    float4 v = smem[(w * PIPE + (t & (PIPE - 1))) * 32 + l];
    unsigned e = baseElem + (unsigned)t * 128u;     // 128 elems per wave-tile
    int x = (int)(e >> 9);        // row   (T_out index), fixed across 4 elems
    int y = (int)(e & 511u);      // col   (T_in index), y..y+3
    float fx  = (float)x * inv_ol;
    bool  xin = x < ol;
    float a[4] = {v.x, v.y, v.z, v.w};
    #pragma unroll
    for (int k = 0; k < 4; ++k) {
      float dy = (float)(y + k) * inv_il - fx;
      float g  = 1.0f - __expf(NEG_INV_2SIG2 * dy * dy);
      float ee = (xin && ((y + k) < il)) ? g * a[k] : 0.0f;
      s1 += ee;
      s2 += ee * ee;
    }
  };

  // Prologue: put PIPE-1 tiles in flight.
  #pragma unroll
  for (int t = 0; t < PIPE - 1; ++t) issue(t);

  // Steady state: issue tile t+3, wait so tile t has landed (in-order
  // completion: <=3 outstanding of 4 issued => oldest done), consume t.
  // Unroll by PIPE so the LDS ring index (t & 3) folds to immediates.
  #pragma unroll 4
  for (int t = 0; t < TILES - (PIPE - 1); ++t) {
    issue(t + PIPE - 1);
    asm volatile("s_wait_asynccnt 3" ::: "memory");
    consume(t);
  }
  // Drain.
  asm volatile("s_wait_asynccnt 2" ::: "memory");
  consume(TILES - 3);
  asm volatile("s_wait_asynccnt 1" ::: "memory");
  consume(TILES - 2);
  asm volatile("s_wait_asynccnt 0" ::: "memory");
  consume(TILES - 1);

  // wave32 tree reduction
  #pragma unroll
  for (int off = 16; off > 0; off >>= 1) {
    s1 += __shfl_down(s1, off, 32);
    s2 += __shfl_down(s2, off, 32);
  }
  if (l == 0) { red[w] = s1; red[WAVES + w] = s2; }
  __syncthreads();
  if (tid == 0) {
    float a1 = 0.0f, a2 = 0.0f;
    #pragma unroll
    for (int i = 0; i < WAVES; ++i) { a1 += red[i]; a2 += red[WAVES + i]; }
    partials[2 * blk]     = a1;   // deterministic: write, never atomic
    partials[2 * blk + 1] = a2;
  }
}

__global__ __launch_bounds__(32) void gal_final_kernel(
    const float* __restrict__ partials,
    const int*   __restrict__ olens,
    float*       __restrict__ out)
{
  const int b = blockIdx.x;     // 64 blocks, 32 threads (one wave)
  const int j = threadIdx.x;
  float p1 = partials[2 * (b * BLK_PER_B + j)];
  float p2 = partials[2 * (b * BLK_PER_B + j) + 1];
  #pragma unroll
  for (int off = 16; off > 0; off >>= 1) {
    p1 += __shfl_down(p1, off, 32);
    p2 += __shfl_down(p2, off, 32);
  }
  if (j == 0) {
    float inv = 1.0f / (float)olens[b];
    out[b]      = p1 * inv;   // l1_mae
    out[NB + b] = p2 * inv;   // l2_mse
  }
}

extern "C" void kernel_launch(void* const* d_in, const int* in_sizes, int n_in,
                              void* d_out, int out_size, void* d_ws, size_t ws_size,
                              hipStream_t stream) {
  (void)in_sizes; (void)n_in; (void)out_size; (void)ws_size;
  const float* att   = (const float*)d_in[0];
  const int*   ilens = (const int*)d_in[1];
  const int*   olens = (const int*)d_in[2];
  float*       ws    = (float*)d_ws;     // 2048 blocks * 2 floats = 16 KiB
  float*       out   = (float*)d_out;    // 128 floats: [l1_mae(64), l2_mse(64)]

  gal_partial_kernel<<<NB * BLK_PER_B, THREADS, 0, stream>>>(att, ilens, olens, ws);
  gal_final_kernel<<<NB, 32, 0, stream>>>(ws, olens, out);
}